// FakeQuantizeBase_90692529423049
// MI455X (gfx1250) — compile-verified
//
#include <hip/hip_runtime.h>

// FakeQuantize (group=128, qmin=-8, qmax=7) for MI455X / gfx1250.
//
// Memory-bound streaming op: 516 MB traffic -> ~22 us floor @ 23.3 TB/s.
// Memory path (verified in round 1 asm): b128 NT loads/stores, scalar
// s_load for wave-uniform scale/zp (one wave32 == one 128-elem group).
//
// This round: cut VALU/element from ~16 to ~7 so the kernel cannot become
// VALU-issue bound:
//  - wave-uniform correctly-rounded reciprocal (rcp + 2 Newton fmas, done
//    once per wave) + 3-op Markstein division per element (== IEEE x/s for
//    our in-range values),
//  - fold zero_point through the clamp:
//    clip(rint(v)+z,-8,7)-z == clamp(rint(v), -8-z, 7-z)  (exact: integer-
//    valued fp32, |values| << 2^24), bounds computed once per wave.

typedef float v4f __attribute__((ext_vector_type(4)));

#define QMIN (-8.0f)
#define QMAX (7.0f)

__global__ __launch_bounds__(256) void fakequant_g128_kernel(
    const float* __restrict__ X,
    const float* __restrict__ scale,
    const int*   __restrict__ zero_point,
    float*       __restrict__ out,
    int nvec)  // number of float4 elements
{
    int idx = blockIdx.x * blockDim.x + threadIdx.x;   // float4 index
    if (idx >= nvec) return;

    // Flat element e = idx*4; group = e/128 = idx>>5; uniform across the
    // wave (blocks 32-aligned) -> scalar loads through the constant cache.
    int sidx = __builtin_amdgcn_readfirstlane(idx >> 5);
    float s = scale[sidx];
    float z = (float)zero_point[sidx];

    // Wave-uniform prep (amortized over 128 elements):
    // correctly-rounded 1/s via v_rcp_f32 + 2 Newton steps (fma).
    float y = __builtin_amdgcn_rcpf(s);
    y = __builtin_fmaf(__builtin_fmaf(-s, y, 1.0f), y, y);
    y = __builtin_fmaf(__builtin_fmaf(-s, y, 1.0f), y, y);
    // Shifted clamp bounds (exact integer fp32 arithmetic).
    float lo = QMIN - z;
    float hi = QMAX - z;

    const v4f* __restrict__ Xv = (const v4f*)X;
    v4f*       __restrict__ Ov = (v4f*)out;

    // X read once, out written once: NT hints keep the 512 MB stream from
    // churning the 192 MB L2.
    v4f x = __builtin_nontemporal_load(Xv + idx);

    v4f r;
#pragma unroll
    for (int i = 0; i < 4; ++i) {
        // Markstein: correctly-rounded x/s in 3 ops (mul + 2 fma).
        float q0 = x[i] * y;
        float rr = __builtin_fmaf(-s, q0, x[i]);
        float q  = __builtin_fmaf(rr, y, q0);      // == IEEE fl(x/s)
        float c  = __builtin_rintf(q);             // v_rndne_f32 (RNE)
        c = fminf(fmaxf(c, lo), hi);               // clamp (v_med3 / max+min)
        r[i] = c * s;
    }

    __builtin_nontemporal_store(r, Ov + idx);
}

extern "C" void kernel_launch(void* const* d_in, const int* in_sizes, int n_in,
                              void* d_out, int out_size, void* d_ws, size_t ws_size,
                              hipStream_t stream)
{
    const float* X     = (const float*)d_in[0];
    const float* scale = (const float*)d_in[1];
    const int*   zp    = (const int*)d_in[2];
    float*       out   = (float*)d_out;

    int n    = in_sizes[0];      // N*D, divisible by 128 (group) and by 4
    int nvec = n >> 2;           // float4 count

    const int block = 256;       // 8 wave32s per block
    int grid = (nvec + block - 1) / block;

    fakequant_g128_kernel<<<grid, block, 0, stream>>>(X, scale, zp, out, nvec);
}